// PCRLoss_78700980732407
// MI455X (gfx1250) — compile-verified
//
#include <hip/hip_runtime.h>

typedef float v2f __attribute__((ext_vector_type(2)));
typedef float v8f __attribute__((ext_vector_type(8)));

namespace {
constexpr int   kB = 16;
constexpr int   kN = 2048;
constexpr float kEps = 0.05f;
constexpr int   kIters = 20;
constexpr int   kRowsPerBlock = 128;            // 8 waves * 16 rows each
constexpr int   kBlocksX = kN / kRowsPerBlock;  // 16
constexpr float kLn2 = 0.69314718055994531f;
// base-2 sharpness: (1/eps) * log2(e)
constexpr float kK2 = 1.4426950408889634f / kEps;
}

// ---------------------------------------------------------------------------
// init: pack points as float4 (x, y, z, |p|^2); zero the Sinkhorn potentials.
// pk layout: [3 sets][B*N] float4  (set 0 = gt, 1 = pred0, 2 = pred1)
// ---------------------------------------------------------------------------
__global__ __launch_bounds__(256) void init_kernel(
    const float* __restrict__ gt, const float* __restrict__ p0,
    const float* __restrict__ p1, float4* __restrict__ pk,
    float* __restrict__ f, float* __restrict__ g) {
  int i = blockIdx.x * blockDim.x + threadIdx.x;
  if (i < 3 * kB * kN) {
    int set = i / (kB * kN);
    int idx = i - set * (kB * kN);
    const float* p = (set == 0) ? gt : ((set == 1) ? p0 : p1);
    float a = p[3 * idx], b = p[3 * idx + 1], c = p[3 * idx + 2];
    pk[i] = make_float4(a, b, c, a * a + b * b + c * c);
  }
  if (i < 2 * kB * kN) {  // f and g: [2 preds][B*N] each
    f[i] = 0.0f;
    g[i] = 0.0f;
  }
}

// ---------------------------------------------------------------------------
// WMMA distance tile (V_WMMA_F32_16X16X4_F32, homogeneous coordinates):
// A row (16x4): lanes<16 K=0,1 = (-2x0,-2x1); lanes>=16 K=2,3 = (-2x2, 1)
// B col (4x16): lanes<16 K=0,1 = (y0,y1);     lanes>=16 K=2,3 = (y2, |y|^2)
// -> C[r][col] + |x_row|^2 = squared distance.
// C/D layout: lane L<16 -> col L rows 0..7 in v[0..7]; lane>=16 -> rows 8..15.
// The packed float4 makes the per-tile B fragment one aligned b64 load:
// granule (2*m + hi) of the float2 view.
// ---------------------------------------------------------------------------

// Online-logsumexp pass (base-2 domain, single v_exp_f32 per element):
// out[b,n] = eps*log(1/N) - eps*LSE_m((pot[b,m]-C[n,m])/eps)
//          = -eps*ln2*(11 + mx + log2(sm))     [since N = 2^11]
__global__ __launch_bounds__(256) void lse_pass_kernel(
    const float4* __restrict__ pkX, const float4* __restrict__ pkY,
    const float* __restrict__ pot, float* __restrict__ out) {
  const int b = blockIdx.y;
  const int lane = threadIdx.x & 31;
  const int wave = threadIdx.x >> 5;
  const int hi = lane >> 4;
  const int l15 = lane & 15;
  const int row0 = blockIdx.x * kRowsPerBlock + wave * 16;

  const float4* xb = pkX + (size_t)b * kN + row0;
  const float2* yb2 = reinterpret_cast<const float2*>(pkY + (size_t)b * kN);
  const float* potb = pot + (size_t)b * kN;

  float4 xp = xb[l15];
  v2f a;
  a.x = hi ? (-2.0f * xp.z) : (-2.0f * xp.x);
  a.y = hi ? 1.0f           : (-2.0f * xp.y);

  float xsq[8];
#pragma unroll
  for (int r = 0; r < 8; ++r) xsq[r] = xb[r + 8 * hi].w;

  float mx[8], sm[8];
#pragma unroll
  for (int r = 0; r < 8; ++r) { mx[r] = -__builtin_inff(); sm[r] = 0.0f; }

#pragma unroll 2
  for (int m0 = 0; m0 < kN; m0 += 16) {
    const int m = m0 + l15;
    float2 bv = yb2[2 * m + hi];  // (y0,y1) or (y2,|y|^2)
    v2f bf;
    bf.x = bv.x;
    bf.y = bv.y;
    v8f c = {};
    c = __builtin_amdgcn_wmma_f32_16x16x4_f32(false, a, false, bf, (short)0, c,
                                              false, false);
    float gK = potb[m] * kK2;
#pragma unroll
    for (int r = 0; r < 8; ++r) {
      float d = fmaxf(c[r] + xsq[r], 0.0f) + 1e-12f;
      float cost = __builtin_amdgcn_sqrtf(d);        // raw v_sqrt_f32
      float t = __builtin_fmaf(-kK2, cost, gK);      // (g - C)/eps in log2
      float nm = fmaxf(mx[r], t);
      float e = __builtin_amdgcn_exp2f(-fabsf(mx[r] - t));  // one trans op
      bool grew = t > mx[r];
      sm[r] = __builtin_fmaf(sm[r], grew ? e : 1.0f, grew ? 1.0f : e);
      mx[r] = nm;
    }
  }

  // merge online-LSE states across the 16 lanes holding different columns
#pragma unroll
  for (int d = 1; d < 16; d <<= 1) {
#pragma unroll
    for (int r = 0; r < 8; ++r) {
      float omx = __shfl_xor(mx[r], d, 32);
      float osm = __shfl_xor(sm[r], d, 32);
      float nm = fmaxf(mx[r], omx);
      float e = __builtin_amdgcn_exp2f(-fabsf(mx[r] - omx));
      bool obig = omx > mx[r];
      sm[r] = __builtin_fmaf(sm[r], obig ? e : 1.0f,
                             osm * (obig ? 1.0f : e));
      mx[r] = nm;
    }
  }

  if (l15 == 0) {
    float* ob = out + (size_t)b * kN + row0;
#pragma unroll
    for (int r = 0; r < 8; ++r) {
      float l2 = mx[r] + __builtin_amdgcn_logf(sm[r]);  // raw v_log_f32 (log2)
      ob[r + 8 * hi] = -kEps * kLn2 * (11.0f + l2);
    }
  }
}

// Chamfer half: partial[block] = sum over 128 rows of min_m ||x_n - y_m||^2
__global__ __launch_bounds__(256) void chamfer_min_kernel(
    const float4* __restrict__ pkX, const float4* __restrict__ pkY,
    float* __restrict__ partial) {
  __shared__ float red[8];
  const int b = blockIdx.y;
  const int lane = threadIdx.x & 31;
  const int wave = threadIdx.x >> 5;
  const int hi = lane >> 4;
  const int l15 = lane & 15;
  const int row0 = blockIdx.x * kRowsPerBlock + wave * 16;

  const float4* xb = pkX + (size_t)b * kN + row0;
  const float2* yb2 = reinterpret_cast<const float2*>(pkY + (size_t)b * kN);

  float4 xp = xb[l15];
  v2f a;
  a.x = hi ? (-2.0f * xp.z) : (-2.0f * xp.x);
  a.y = hi ? 1.0f           : (-2.0f * xp.y);

  float xsq[8];
#pragma unroll
  for (int r = 0; r < 8; ++r) xsq[r] = xb[r + 8 * hi].w;

  float mn[8];
#pragma unroll
  for (int r = 0; r < 8; ++r) mn[r] = __builtin_inff();

#pragma unroll 2
  for (int m0 = 0; m0 < kN; m0 += 16) {
    const int m = m0 + l15;
    float2 bv = yb2[2 * m + hi];
    v2f bf;
    bf.x = bv.x;
    bf.y = bv.y;
    v8f c = {};
    c = __builtin_amdgcn_wmma_f32_16x16x4_f32(false, a, false, bf, (short)0, c,
                                              false, false);
#pragma unroll
    for (int r = 0; r < 8; ++r)
      mn[r] = fminf(mn[r], fmaxf(c[r] + xsq[r], 0.0f));
  }

#pragma unroll
  for (int d = 1; d < 16; d <<= 1) {
#pragma unroll
    for (int r = 0; r < 8; ++r) mn[r] = fminf(mn[r], __shfl_xor(mn[r], d, 32));
  }

  float s = 0.0f;
#pragma unroll
  for (int r = 0; r < 8; ++r) s += mn[r];
  s += __shfl_xor(s, 16, 32);  // combine rows 0..7 and 8..15 halves
  if (lane == 0) red[wave] = s;
  __syncthreads();
  if (threadIdx.x == 0) {
    float t = 0.0f;
#pragma unroll
    for (int w = 0; w < 8; ++w) t += red[w];
    partial[blockIdx.y * gridDim.x + blockIdx.x] = t;
  }
}

// Transport cost: partial[block] = sum exp2((f[n]+g[m]-C)*K2) * C over rows
__global__ __launch_bounds__(256) void emd_cost_kernel(
    const float4* __restrict__ pkX, const float4* __restrict__ pkY,
    const float* __restrict__ fpot, const float* __restrict__ gpot,
    float* __restrict__ partial) {
  __shared__ float red[8];
  const int b = blockIdx.y;
  const int lane = threadIdx.x & 31;
  const int wave = threadIdx.x >> 5;
  const int hi = lane >> 4;
  const int l15 = lane & 15;
  const int row0 = blockIdx.x * kRowsPerBlock + wave * 16;

  const float4* xb = pkX + (size_t)b * kN + row0;
  const float2* yb2 = reinterpret_cast<const float2*>(pkY + (size_t)b * kN);
  const float* fb = fpot + (size_t)b * kN + row0;
  const float* gb = gpot + (size_t)b * kN;

  float4 xp = xb[l15];
  v2f a;
  a.x = hi ? (-2.0f * xp.z) : (-2.0f * xp.x);
  a.y = hi ? 1.0f           : (-2.0f * xp.y);

  float xsq[8], frowK[8];
#pragma unroll
  for (int r = 0; r < 8; ++r) {
    xsq[r] = xb[r + 8 * hi].w;
    frowK[r] = fb[r + 8 * hi] * kK2;
  }

  float acc = 0.0f;
#pragma unroll 2
  for (int m0 = 0; m0 < kN; m0 += 16) {
    const int m = m0 + l15;
    float2 bv = yb2[2 * m + hi];
    v2f bf;
    bf.x = bv.x;
    bf.y = bv.y;
    v8f c = {};
    c = __builtin_amdgcn_wmma_f32_16x16x4_f32(false, a, false, bf, (short)0, c,
                                              false, false);
    float gK = gb[m] * kK2;
#pragma unroll
    for (int r = 0; r < 8; ++r) {
      float d = fmaxf(c[r] + xsq[r], 0.0f) + 1e-12f;
      float cost = __builtin_amdgcn_sqrtf(d);
      float t = __builtin_fmaf(-kK2, cost, frowK[r] + gK);
      acc = __builtin_fmaf(__builtin_amdgcn_exp2f(t), cost, acc);
    }
  }

#pragma unroll
  for (int d = 1; d < 32; d <<= 1) acc += __shfl_xor(acc, d, 32);
  if (lane == 0) red[wave] = acc;
  __syncthreads();
  if (threadIdx.x == 0) {
    float t = 0.0f;
#pragma unroll
    for (int w = 0; w < 8; ++w) t += red[w];
    partial[blockIdx.y * gridDim.x + blockIdx.x] = t;
  }
}

// Deterministic fixed-order final reduction:
// loss = sum(chamfer partials)/(B*N) + sum(emd partials)/B
__global__ __launch_bounds__(256) void reduce_kernel(
    const float* __restrict__ chp, const float* __restrict__ emp,
    float* __restrict__ out) {
  __shared__ float sh[256];
  int t = threadIdx.x;
  float a = 0.0f;
  for (int i = t; i < 4 * kB * kBlocksX; i += 256) a += chp[i];
  a *= 1.0f / (float)(kB * kN);
  float e = 0.0f;
  for (int i = t; i < 2 * kB * kBlocksX; i += 256) e += emp[i];
  e *= 1.0f / (float)kB;
  sh[t] = a + e;
  __syncthreads();
  for (int s = 128; s > 0; s >>= 1) {
    if (t < s) sh[t] += sh[t + s];
    __syncthreads();
  }
  if (t == 0) out[0] = sh[0];
}

// ---------------------------------------------------------------------------
extern "C" void kernel_launch(void* const* d_in, const int* in_sizes, int n_in,
                              void* d_out, int out_size, void* d_ws,
                              size_t ws_size, hipStream_t stream) {
  const float* gt = (const float*)d_in[0];
  // d_in[1] = gt_normals (unused by the reference loss)
  const float* p0 = (const float*)d_in[2];
  const float* p1 = (const float*)d_in[3];
  float* out = (float*)d_out;

  float* ws = (float*)d_ws;
  float* f = ws;                           // [2][B*N]
  float* g = ws + 2 * kB * kN;             // [2][B*N]
  float4* pk = (float4*)(ws + 4 * kB * kN);  // [3][B*N] float4 (16B aligned)
  float* chp = ws + 4 * kB * kN + 3 * kB * kN * 4;  // 4*256 chamfer partials
  float* emp = chp + 4 * kB * kBlocksX;              // 2*256 emd partials

  dim3 grid(kBlocksX, kB);
  dim3 blk(256);

  init_kernel<<<dim3((3 * kB * kN + 255) / 256), blk, 0, stream>>>(gt, p0, p1,
                                                                   pk, f, g);

  for (int p = 0; p < 2; ++p) {
    const float4* pkG = pk;                        // gt packed
    const float4* pkP = pk + (1 + p) * kB * kN;    // pred packed
    float* fp = f + p * kB * kN;
    float* gp = g + p * kB * kN;

    // chamfer_loss(gt, P): min over P-cols per gt-row + min over gt-cols per P-row
    chamfer_min_kernel<<<grid, blk, 0, stream>>>(
        pkG, pkP, chp + (2 * p + 0) * kB * kBlocksX);
    chamfer_min_kernel<<<grid, blk, 0, stream>>>(
        pkP, pkG, chp + (2 * p + 1) * kB * kBlocksX);

    // Sinkhorn: emd_loss(P, gt) -> rows = P (f), cols = gt (g)
    for (int it = 0; it < kIters; ++it) {
      lse_pass_kernel<<<grid, blk, 0, stream>>>(pkP, pkG, gp, fp);
      lse_pass_kernel<<<grid, blk, 0, stream>>>(pkG, pkP, fp, gp);
    }
    emd_cost_kernel<<<grid, blk, 0, stream>>>(pkP, pkG, fp, gp,
                                              emp + p * kB * kBlocksX);
  }

  reduce_kernel<<<dim3(1), blk, 0, stream>>>(chp, emp, out);
}